// FlowWarp_7748121002235
// MI455X (gfx1250) — compile-verified
//
#include <hip/hip_runtime.h>

typedef __attribute__((ext_vector_type(2))) float v2f;
typedef __attribute__((ext_vector_type(4))) float v4f;
typedef __attribute__((ext_vector_type(8))) float v8f;
typedef int v4i_ __attribute__((vector_size(16)));   // matches async-LDS builtin pointee

#define TILE_PIX 128   // pixels per workgroup (8 waves x 16 pixels)
#define BLOCK    256
#define KDIM     130
#define KPAD     132   // pad K to multiple of 4; 132-float row skew avoids LDS bank conflicts
#define FOUT     64

#if defined(__HIP_DEVICE_COMPILE__) && defined(__has_builtin)
#  if __has_builtin(__builtin_amdgcn_global_load_async_to_lds_b128)
#    define ASYNC_XC 1
#  else
#    define ASYNC_XC 0
#  endif
#  if __has_builtin(__builtin_amdgcn_s_wait_asynccnt)
#    define WAIT_ASYNC() __builtin_amdgcn_s_wait_asynccnt(0)
#  else
#    define WAIT_ASYNC() asm volatile("s_wait_asynccnt 0" ::: "memory")
#  endif
#else
#  define ASYNC_XC 0
#  define WAIT_ASYNC() ((void)0)
#endif

template <int H, int W, bool FULL>
__global__ __launch_bounds__(BLOCK) void flowwarp_level_kernel(
    const float* __restrict__ xc,
    const float* __restrict__ xp,
    const float* __restrict__ fl,
    const float* __restrict__ dw,
    const float* __restrict__ pw,
    const float* __restrict__ bias,
    float* __restrict__ out)
{
    constexpr int HW   = H * W;
    constexpr int NPIX = 4 * H * W;
    constexpr bool kAsync = FULL && (ASYNC_XC != 0);

    extern __shared__ float smem[];
    float (*sX)[KPAD] = (float (*)[KPAD])smem;                       // [128][132]
    float (*sW)[KPAD] = (float (*)[KPAD])(smem + TILE_PIX * KPAD);   // [64][132] transposed W'
    float* sB = smem + TILE_PIX * KPAD + FOUT * KPAD;                // [64]

    const int tid  = threadIdx.x;
    const int lane = tid & 31;
    const int wave = tid >> 5;
    const int pixBase = blockIdx.x * TILE_PIX;
    const int half = lane >> 4;          // half-wave -> which of 2 pixels per iter
    const int ch4  = (lane & 15) * 4;    // 4 consecutive channels per lane

    // ---- Phase 0 (CDNA5 async path): xc tile -> LDS via ASYNC VMEM->LDS (ASYNCcnt).
    // 2 pixels per issue: 32 lanes x 16B = 512B. Each wave fills only its own rows,
    // so a single s_wait_asynccnt 0 before its WMMA reads is sufficient ordering.
    if constexpr (kAsync) {
        #pragma unroll
        for (int pi2 = 0; pi2 < 16; pi2 += 2) {
            const int p  = wave * 16 + pi2 + half;
            const int gp = pixBase + p;
            const float* g = xc + (size_t)gp * 64 + ch4;
            float* l = &sX[p][ch4];
#if ASYNC_XC
            __builtin_amdgcn_global_load_async_to_lds_b128(
                (__attribute__((address_space(1))) v4i_*)g,
                (__attribute__((address_space(3))) v4i_*)l,
                0, 0);
#endif
        }
    }

    // ---- Phase 1: fold dw into pw, store transposed sW[f][c]; pad rows >=130 with 0
    for (int idx = tid; idx < FOUT * KPAD; idx += BLOCK) {
        const int f = idx / KPAD;
        const int c = idx - f * KPAD;
        float v = 0.0f;
        if (c < KDIM) v = dw[c] * pw[c * FOUT + f];
        sW[f][c] = v;
    }
    if (tid < FOUT) sB[tid] = bias[tid];

    // ---- Phase 2: warped features + flow channels (+ xc copy when not async).
    // 2 pixels per iteration (one per half-wave); each lane gathers 4 consecutive
    // channels (b128) from each bilinear corner -> 4 wide coalesced loads / 2 pixels.
    for (int pi2 = 0; pi2 < 16; pi2 += 2) {
        const int p  = wave * 16 + pi2 + half;
        const int gp = pixBase + p;
        const bool valid = FULL || (gp < NPIX);

        int b = 0, y = 0, x = 0;
        float fy = 0.0f, fx = 0.0f;
        if (valid) {
            b = gp / HW;                 // compile-time HW -> magic multiply
            const int rem = gp - b * HW;
            y = rem / W;
            x = rem - y * W;
            fy = fl[(size_t)gp * 2 + 0];
            fx = fl[(size_t)gp * 2 + 1];
        }
        // query = grid + fl  (dense_image_warp called with flow = -fl)
        const float qy = (float)y + fy;
        const float qx = (float)x + fx;
        const float y0f = fminf(fmaxf(floorf(qy), 0.0f), (float)(H - 2));
        const float x0f = fminf(fmaxf(floorf(qx), 0.0f), (float)(W - 2));
        const int   y0  = (int)y0f;
        const int   x0  = (int)x0f;
        const float ay  = fminf(fmaxf(qy - y0f, 0.0f), 1.0f);
        const float ax  = fminf(fmaxf(qx - x0f, 0.0f), 1.0f);

        const float* basep = xp + (size_t)b * HW * 64;
        const float* ptl = basep + ((size_t)y0 * W + x0) * 64 + ch4;

        v4f wv = {};
        if (valid) {
            const v4f tl = *(const v4f*)(ptl);
            const v4f tr = *(const v4f*)(ptl + 64);
            const v4f bl = *(const v4f*)(ptl + (size_t)W * 64);
            const v4f br = *(const v4f*)(ptl + (size_t)W * 64 + 64);
            const v4f top = tl + (tr - tl) * ax;
            const v4f bot = bl + (br - bl) * ax;
            wv = top + (bot - top) * ay;
        }
        *(v4f*)&sX[p][64 + ch4] = wv;

        if constexpr (!kAsync) {
            v4f xcv = {};
            if (valid) xcv = *(const v4f*)(xc + (size_t)gp * 64 + ch4);
            *(v4f*)&sX[p][ch4] = xcv;
        }

        if ((lane & 15) == 0) {
            sX[p][128] = valid ? fy : 0.0f;
            sX[p][129] = valid ? fx : 0.0f;
            sX[p][130] = 0.0f;
            sX[p][131] = 0.0f;
        }
    }
    if constexpr (kAsync) WAIT_ASYNC();
    __syncthreads();

    // ---- Phase 3: WMMA f32 16x16x4. Wave owns 16 pixels x 64 outputs (4 N-tiles).
    const int m = lane & 15;     // A row / B column within tile
    const int h = lane >> 4;     // half-wave selects K pair {2h, 2h+1}
    const int rowBase = wave * 16;

    v8f acc0 = {}, acc1 = {}, acc2 = {}, acc3 = {};
    for (int kb = 0; kb < KPAD; kb += 4) {
        const int k0 = kb + 2 * h;
        const v2f a  = *(const v2f*)&sX[rowBase + m][k0];
        const v2f b0 = *(const v2f*)&sW[m +  0][k0];
        const v2f b1 = *(const v2f*)&sW[m + 16][k0];
        const v2f b2 = *(const v2f*)&sW[m + 32][k0];
        const v2f b3 = *(const v2f*)&sW[m + 48][k0];
        acc0 = __builtin_amdgcn_wmma_f32_16x16x4_f32(false, a, false, b0, (short)0, acc0, false, false);
        acc1 = __builtin_amdgcn_wmma_f32_16x16x4_f32(false, a, false, b1, (short)0, acc1, false, false);
        acc2 = __builtin_amdgcn_wmma_f32_16x16x4_f32(false, a, false, b2, (short)0, acc2, false, false);
        acc3 = __builtin_amdgcn_wmma_f32_16x16x4_f32(false, a, false, b3, (short)0, acc3, false, false);
    }

    // ---- Epilogue: D[r + 8h][n], n = lane&15 ; add bias, store NHWC
    const float b0v = sB[m +  0];
    const float b1v = sB[m + 16];
    const float b2v = sB[m + 32];
    const float b3v = sB[m + 48];
    #pragma unroll
    for (int r = 0; r < 8; ++r) {
        const int p  = rowBase + r + 8 * h;
        const int gp = pixBase + p;
        if (FULL || gp < NPIX) {
            float* o = out + (size_t)gp * 64 + m;
            o[0]  = acc0[r] + b0v;
            o[16] = acc1[r] + b1v;
            o[32] = acc2[r] + b2v;
            o[48] = acc3[r] + b3v;
        }
    }
}

template <int H, int W, bool FULL>
static inline size_t launch_level(void* const* d_in, int i, float* out, size_t off,
                                  hipStream_t stream)
{
    const float* xc   = (const float*)d_in[6 * i + 0];
    const float* xp   = (const float*)d_in[6 * i + 1];
    const float* fl   = (const float*)d_in[6 * i + 2];
    const float* dw   = (const float*)d_in[6 * i + 3];
    const float* pw   = (const float*)d_in[6 * i + 4];
    const float* bias = (const float*)d_in[6 * i + 5];
    constexpr int numPix = 4 * H * W;
    constexpr int grid = (numPix + TILE_PIX - 1) / TILE_PIX;
    const size_t shmem = (size_t)(TILE_PIX * KPAD + FOUT * KPAD + FOUT) * sizeof(float);
    flowwarp_level_kernel<H, W, FULL><<<grid, BLOCK, shmem, stream>>>(
        xc, xp, fl, dw, pw, bias, out + off);
    return off + (size_t)numPix * FOUT;
}

extern "C" void kernel_launch(void* const* d_in, const int* in_sizes, int n_in,
                              void* d_out, int out_size, void* d_ws, size_t ws_size,
                              hipStream_t stream) {
    (void)in_sizes; (void)n_in; (void)out_size; (void)d_ws; (void)ws_size;
    float* out = (float*)d_out;
    size_t off = 0;
    off = launch_level<192, 384, true >(d_in, 0, out, off, stream);  // 294912 px
    off = launch_level< 96, 192, true >(d_in, 1, out, off, stream);  //  73728 px
    off = launch_level< 48,  96, true >(d_in, 2, out, off, stream);  //  18432 px
    off = launch_level< 24,  48, true >(d_in, 3, out, off, stream);  //   4608 px
    off = launch_level< 12,  24, true >(d_in, 4, out, off, stream);  //   1152 px
    off = launch_level<  6,  12, false>(d_in, 5, out, off, stream);  //    288 px (tail)
}